// GNN_79379585564874
// MI455X (gfx1250) — compile-verified
//
#include <hip/hip_runtime.h>
#include <hip/hip_bf16.h>
#include <math.h>

typedef __attribute__((ext_vector_type(2))) float v2f;
typedef __attribute__((ext_vector_type(8))) float v8f;

#define N_NODES 50000
#define IN_DIM  128
#define HID_DIM 128
#define OUT_DIM 64
#define N_EDGES 800000

// ---------------------------------------------------------------------------
// Kernel 1: h = relu(x @ W1 + b1)
// grid.x = N_NODES/16 blocks; block = 256 threads = 8 waves.
// Wave w computes the 16x16 output tile at columns [16w, 16w+16).
// f32 WMMA: A = 16x4 (2 VGPRs), B = 4x16 (2 VGPRs), C/D = 16x16 (8 VGPRs).
// A layout: lane L holds M = L&15; VGPR0/1 = K = {k+2*(L>>4), k+2*(L>>4)+1}.
// B layout: lane L holds N = L&15; VGPR0/1 = K rows {2*(L>>4), 2*(L>>4)+1}.
// C layout: VGPR r, lane L -> (M = r + 8*(L>>4), N = L&15).
// ---------------------------------------------------------------------------
__global__ __launch_bounds__(256) void k_linear1_relu(
    const float* __restrict__ x, const float* __restrict__ W1,
    const float* __restrict__ b1, float* __restrict__ h)
{
  const int tid  = threadIdx.x;
  const int wave = tid >> 5;
  const int lane = tid & 31;
  const int m0   = blockIdx.x * 16;
  const int n0   = wave * 16;
  const int mrow = lane & 15;
  const int hi   = lane >> 4;   // selects K pair {0,1} vs {2,3}
  const int ncol = lane & 15;

  v8f acc = {};
  const float* xrow = x + (size_t)(m0 + mrow) * IN_DIM + 2 * hi;
  const float* wcol = W1 + n0 + ncol + (size_t)(2 * hi) * HID_DIM;

  #pragma unroll 4
  for (int k = 0; k < IN_DIM; k += 4) {
    v2f a, b;
    a.x = xrow[k];
    a.y = xrow[k + 1];
    b.x = wcol[(size_t)k * HID_DIM];
    b.y = wcol[(size_t)(k + 1) * HID_DIM];
    acc = __builtin_amdgcn_wmma_f32_16x16x4_f32(
        /*neg_a=*/false, a, /*neg_b=*/false, b,
        /*c_mod=*/(short)0, acc, /*reuse_a=*/false, /*reuse_b=*/false);
  }

  const float bias = b1[n0 + ncol];
  #pragma unroll
  for (int r = 0; r < 8; ++r) {
    const int m = m0 + r + 8 * hi;
    float v = acc[r] + bias;
    h[(size_t)m * HID_DIM + n0 + ncol] = v > 0.f ? v : 0.f;
  }
}

// ---------------------------------------------------------------------------
// Kernel 2a: zero the aggregation buffer (float4 stores).
// ---------------------------------------------------------------------------
__global__ __launch_bounds__(256) void k_zero4(float4* __restrict__ p, int n4)
{
  int i = blockIdx.x * blockDim.x + threadIdx.x;
  if (i < n4) p[i] = make_float4(0.f, 0.f, 0.f, 0.f);
}

// ---------------------------------------------------------------------------
// Kernel 2b: agg[dst] += h[src] for every edge.
// One wave per edge; lane carries a float4 slice of the 128-float row
// (512 contiguous bytes per edge -> full cachelines; h & agg are L2-resident:
// 2 x 25.6 MB inside the 192 MB L2, so the f32 atomics resolve in L2).
// ---------------------------------------------------------------------------
__global__ __launch_bounds__(256) void k_scatter_add(
    const int* __restrict__ esrc, const int* __restrict__ edst,
    const float* __restrict__ h, float* __restrict__ agg, int n_edges)
{
  const int eid  = blockIdx.x * 8 + (threadIdx.x >> 5);
  const int lane = threadIdx.x & 31;
  if (eid >= n_edges) return;
  const int s = esrc[eid];
  const int d = edst[eid];
  const float4 v = ((const float4*)(h + (size_t)s * HID_DIM))[lane];
  float* ap = agg + (size_t)d * HID_DIM + lane * 4;
  atomicAdd(ap + 0, v.x);
  atomicAdd(ap + 1, v.y);
  atomicAdd(ap + 2, v.z);
  atomicAdd(ap + 3, v.w);
}

// ---------------------------------------------------------------------------
// Kernel 3: out = log_softmax(agg @ W2 + b2)
// Block = 128 threads = 4 waves; block computes a 16x64 logits tile (wave w
// owns N-tile [16w,16w+16)), spills to LDS, does the row-wise max/LSE with 16
// threads, then all 128 threads write back coalesced.
// ---------------------------------------------------------------------------
__global__ __launch_bounds__(128) void k_linear2_logsoftmax(
    const float* __restrict__ agg, const float* __restrict__ W2,
    const float* __restrict__ b2, float* __restrict__ out)
{
  __shared__ float lds[16 * OUT_DIM];
  __shared__ float rowmax[16];
  __shared__ float rowlse[16];

  const int tid  = threadIdx.x;
  const int wave = tid >> 5;
  const int lane = tid & 31;
  const int m0   = blockIdx.x * 16;
  const int n0   = wave * 16;
  const int mrow = lane & 15;
  const int hi   = lane >> 4;
  const int ncol = lane & 15;

  v8f acc = {};
  const float* arow = agg + (size_t)(m0 + mrow) * HID_DIM + 2 * hi;
  const float* wcol = W2 + n0 + ncol + (size_t)(2 * hi) * OUT_DIM;

  #pragma unroll 4
  for (int k = 0; k < HID_DIM; k += 4) {
    v2f a, b;
    a.x = arow[k];
    a.y = arow[k + 1];
    b.x = wcol[(size_t)k * OUT_DIM];
    b.y = wcol[(size_t)(k + 1) * OUT_DIM];
    acc = __builtin_amdgcn_wmma_f32_16x16x4_f32(
        false, a, false, b, (short)0, acc, false, false);
  }

  const float bias = b2[n0 + ncol];
  #pragma unroll
  for (int r = 0; r < 8; ++r)
    lds[(r + 8 * hi) * OUT_DIM + n0 + ncol] = acc[r] + bias;
  __syncthreads();

  if (tid < 16) {
    const float* row = lds + tid * OUT_DIM;
    float mx = -INFINITY;
    for (int c = 0; c < OUT_DIM; ++c) mx = fmaxf(mx, row[c]);
    float s = 0.f;
    for (int c = 0; c < OUT_DIM; ++c) s += expf(row[c] - mx);
    rowmax[tid] = mx;
    rowlse[tid] = logf(s);
  }
  __syncthreads();

  #pragma unroll
  for (int j = 0; j < 8; ++j) {
    const int flat = j * 128 + tid;       // 0 .. 1023
    const int m = flat >> 6;              // row within tile
    out[(size_t)m0 * OUT_DIM + flat] = lds[flat] - rowmax[m] - rowlse[m];
  }
}

// ---------------------------------------------------------------------------
extern "C" void kernel_launch(void* const* d_in, const int* in_sizes, int n_in,
                              void* d_out, int out_size, void* d_ws, size_t ws_size,
                              hipStream_t stream) {
  const float* x   = (const float*)d_in[0];
  const float* W1  = (const float*)d_in[1];
  const float* b1  = (const float*)d_in[2];
  const float* W2  = (const float*)d_in[3];
  const float* b2  = (const float*)d_in[4];
  const int* esrc  = (const int*)d_in[5];
  const int* edst  = (const int*)d_in[6];
  float* out = (float*)d_out;

  float* h   = (float*)d_ws;                        // 50000*128 f32 = 25.6 MB
  float* agg = h + (size_t)N_NODES * HID_DIM;       // another 25.6 MB

  // Phase 1: h = relu(x @ W1 + b1)
  k_linear1_relu<<<N_NODES / 16, 256, 0, stream>>>(x, W1, b1, h);

  // Phase 2: agg = segment_sum over edges
  const int n4 = (N_NODES * HID_DIM) / 4;
  k_zero4<<<(n4 + 255) / 256, 256, 0, stream>>>((float4*)agg, n4);
  k_scatter_add<<<(N_EDGES + 7) / 8, 256, 0, stream>>>(esrc, edst, h, agg, N_EDGES);

  // Phase 3: out = log_softmax(agg @ W2 + b2)
  k_linear2_logsoftmax<<<N_NODES / 16, 128, 0, stream>>>(agg, W2, b2, out);
}